// Decimation_8744553415094
// MI455X (gfx1250) — compile-verified
//
#include <hip/hip_runtime.h>

// Pixel-unshuffle (space-to-depth, factor 2):
//   out[b, 4c + 2*dh + dw, h', w'] = in[b, c, 2h'+dh, 2w'+dw]
// Shapes: in (8, 32, 512, 512) f32 -> out (8, 128, 256, 256) f32.
//
// Pure permutation: 256 MiB read + 256 MiB written, 0 FLOPs.
// MI455X roofline: 512 MiB / 23.3 TB/s ~= 22 us. Memory-bound, so the kernel
// is built for bandwidth:
//  - one 16B global_load_b128 per lane (wave reads 512 contiguous bytes)
//  - two 8B global_store_b64 per lane (each wave store is a 256B contiguous
//    run inside a single output row)
//  - non-temporal cache hints (gfx1250 TH=NT) on both sides: the 512 MiB
//    stream exceeds the 192 MB L2 and every byte is touched exactly once.

// Native clang vector types (required by __builtin_nontemporal_*; HIP's
// float4/float2 are wrapper classes the builtin rejects).
typedef float v4f __attribute__((ext_vector_type(4)));
typedef float v2f __attribute__((ext_vector_type(2)));

namespace {
constexpr unsigned Bn = 8;
constexpr unsigned Cn = 32;
constexpr unsigned Hn = 512;
constexpr unsigned Wn = 512;
constexpr unsigned WQ = Wn / 4;          // 128 float4-groups per input row
constexpr unsigned NTHREADS = Bn * Cn * Hn * WQ;   // 16,777,216
constexpr unsigned OUT_HW = (Hn / 2) * (Wn / 2);   // 65536 floats per out channel
}  // namespace

__global__ __launch_bounds__(256) void decimation_s2d_kernel(
    const v4f* __restrict__ in4, v2f* __restrict__ out2) {
  const unsigned tid = blockIdx.x * blockDim.x + threadIdx.x;  // < NTHREADS

  // tid = ((bc * Hn) + h) * WQ + wq   (wq fastest -> coalesced everywhere)
  const unsigned wq = tid & (WQ - 1);      // input float4-group in row: 0..127
  const unsigned t  = tid >> 7;            // bc * Hn + h
  const unsigned h  = t & (Hn - 1);        // input row 0..511
  const unsigned bc = t >> 9;              // b * Cn + c, 0..255

  // Input is laid out so that in4[tid] is exactly this thread's 4 floats:
  // flat float index = (bc*Hn + h)*Wn + 4*wq = 16B * tid.
  const v4f v = __builtin_nontemporal_load(&in4[tid]);

  const unsigned dh = h & 1u;
  const unsigned hp = h >> 1;              // output row 0..255
  // Output channel (within the flat 4C=128 channel dim), batch folded in:
  //   ch = b*128 + c*4 + 2*dh + dw   with  bc = b*32 + c  ->  bc*4 = b*128 + c*4
  const unsigned ch_even = (bc << 2) + (dh << 1);      // dw = 0
  // float2-unit offset: ((ch * 256) + hp) * 128 + wq
  const unsigned o_even = (ch_even * (Hn / 2) + hp) * (Wn / 4) + wq;
  const unsigned o_odd  = o_even + (OUT_HW >> 1);      // +1 channel = +32768 float2

  // Evens (cols 4wq, 4wq+2) -> out cols (2wq, 2wq+1) of the dw=0 channel.
  const v2f ev = {v.x, v.z};
  __builtin_nontemporal_store(ev, &out2[o_even]);
  // Odds  (cols 4wq+1, 4wq+3) -> out cols (2wq, 2wq+1) of the dw=1 channel.
  const v2f od = {v.y, v.w};
  __builtin_nontemporal_store(od, &out2[o_odd]);
}

extern "C" void kernel_launch(void* const* d_in, const int* in_sizes, int n_in,
                              void* d_out, int out_size, void* d_ws, size_t ws_size,
                              hipStream_t stream) {
  (void)in_sizes; (void)n_in; (void)out_size; (void)d_ws; (void)ws_size;
  const v4f* in4 = reinterpret_cast<const v4f*>(d_in[0]);
  v2f* out2 = reinterpret_cast<v2f*>(d_out);

  constexpr unsigned block = 256;                 // 8 wave32s
  constexpr unsigned grid = NTHREADS / block;     // 65536 blocks, exact
  decimation_s2d_kernel<<<grid, block, 0, stream>>>(in4, out2);
}